// Dummy_Net_16655883174496
// MI455X (gfx1250) — compile-verified
//
#include <hip/hip_runtime.h>
#include <math.h>

#define N_NODES 50000
#define N_EDGES 200000
#define F_INDIM 128
#define H_HEADS 4
#define C_DIM   128
#define HC      512
#define EDGE_DIM 11
#define OUT_DIM 128
#define BN_EPS  1e-5f

typedef float v2f __attribute__((ext_vector_type(2)));
typedef float v8f __attribute__((ext_vector_type(8)));

// ---- order-preserving float <-> uint key (for atomic max on floats) ----
__device__ __forceinline__ unsigned f2key(float f) {
    unsigned b = __float_as_uint(f);
    return (b & 0x80000000u) ? ~b : (b | 0x80000000u);
}
__device__ __forceinline__ float key2f(unsigned k) {
    unsigned b = (k & 0x80000000u) ? (k & 0x7fffffffu) : ~k;
    return __uint_as_float(b);
}

// ============================================================
// fp32 GEMM: C[M,NW] = A[M,K] @ W[K,NW] + bias (opt. ReLU)
// K, NW compile-time -> all B fetches become base+immediate offsets.
// One wave -> 16x64 tile via V_WMMA_F32_16X16X4_F32.
// A 16x4 layout : lanes 0-15 rows, VGPR0={K0,K2}, VGPR1={K1,K3} by lane-half
// B 4x16 layout : lanes 0-15 cols, VGPR0={K0,K2}, VGPR1={K1,K3} by lane-half
// C/D layout    : VGPR r -> row r (lanes 0-15) / row r+8 (lanes 16-31)
// ============================================================
template <int K, int NW, bool RELU>
__global__ void wmma_gemm_bias(const float* __restrict__ A, const float* __restrict__ W,
                               const float* __restrict__ bias, float* __restrict__ Cout,
                               int M)
{
    const int lane = threadIdx.x & 31;
    const int wave = threadIdx.x >> 5;
    const int half = lane >> 4;
    const int l16  = lane & 15;
    const int mtile = blockIdx.y * 4 + wave;
    if (mtile * 16 >= M) return;               // wave-uniform: whole wave exits
    const int row0 = mtile * 16;
    const int col0 = blockIdx.x * 64;

    const float* Ap = A + (size_t)(row0 + l16) * K + 2 * half;     // 8B-aligned pair
    const float* Wp = W + (size_t)(2 * half) * NW + col0 + l16;

    v8f acc0 = {}, acc1 = {}, acc2 = {}, acc3 = {};
#pragma unroll 4
    for (int k0 = 0; k0 < K; k0 += 4) {
        const v2f a = *(const v2f*)Ap;                             // global_load_b64
        v2f b0, b1, b2, b3;
        b0.x = Wp[0];   b0.y = Wp[NW];
        b1.x = Wp[16];  b1.y = Wp[NW + 16];
        b2.x = Wp[32];  b2.y = Wp[NW + 32];
        b3.x = Wp[48];  b3.y = Wp[NW + 48];
        acc0 = __builtin_amdgcn_wmma_f32_16x16x4_f32(false, a, false, b0, (short)0, acc0, false, false);
        acc1 = __builtin_amdgcn_wmma_f32_16x16x4_f32(false, a, false, b1, (short)0, acc1, false, false);
        acc2 = __builtin_amdgcn_wmma_f32_16x16x4_f32(false, a, false, b2, (short)0, acc2, false, false);
        acc3 = __builtin_amdgcn_wmma_f32_16x16x4_f32(false, a, false, b3, (short)0, acc3, false, false);
        Ap += 4;
        Wp += 4 * NW;
    }

    float* Cp = Cout + (size_t)(row0 + 8 * half) * NW + col0 + l16;
#pragma unroll
    for (int s = 0; s < 4; ++s) {
        const v8f acc = (s == 0) ? acc0 : (s == 1) ? acc1 : (s == 2) ? acc2 : acc3;
        const float bv = bias[col0 + s * 16 + l16];
#pragma unroll
        for (int rr = 0; rr < 8; ++rr) {
            float vv = acc[rr] + bv;
            if (RELU) vv = fmaxf(vv, 0.f);
            Cp[(size_t)rr * NW + s * 16] = vv;
        }
    }
}

// ---- on-the-fly edge embedding: 4 consecutive channels of (edge_attr @ We + be) ----
__device__ __forceinline__ float4 edge_feat4(const float* __restrict__ ea,
                                             const float* __restrict__ We,
                                             const float* __restrict__ be,
                                             int col)
{
    float4 acc = *(const float4*)(be + col);
#pragma unroll
    for (int j = 0; j < EDGE_DIM; ++j) {
        const float aj = ea[j];
        const float4 w = *(const float4*)(We + (size_t)j * HC + col);
        acc.x += aj * w.x; acc.y += aj * w.y;
        acc.z += aj * w.z; acc.w += aj * w.w;
    }
    return acc;
}

// alpha[e,h] = <q[dst,h,:], k[src,h,:]+e[e,h,:]> / sqrt(C) ; running segment max
__global__ void edge_alpha_kernel(const float* __restrict__ q, const float* __restrict__ k,
                                  const float* __restrict__ edge_attr,
                                  const float* __restrict__ We, const float* __restrict__ be,
                                  const int* __restrict__ ei,
                                  float* __restrict__ alpha, unsigned* __restrict__ amaxk)
{
    const int eid  = blockIdx.x;
    const int h    = threadIdx.x >> 5;
    const int lane = threadIdx.x & 31;
    const int src = ei[eid];
    const int dst = ei[N_EDGES + eid];
    const int col = h * C_DIM + lane * 4;
    const float4 qv = *(const float4*)(q + (size_t)dst * HC + col);
    const float4 kv = *(const float4*)(k + (size_t)src * HC + col);
    const float4 ev = edge_feat4(edge_attr + (size_t)eid * EDGE_DIM, We, be, col);
    float d = qv.x*(kv.x+ev.x) + qv.y*(kv.y+ev.y) + qv.z*(kv.z+ev.z) + qv.w*(kv.w+ev.w);
#pragma unroll
    for (int m = 16; m > 0; m >>= 1) d += __shfl_xor(d, m);
    if (lane == 0) {
        const float a = d * 0.08838834764831845f;   // 1/sqrt(128)
        alpha[(size_t)eid * H_HEADS + h] = a;
        atomicMax(&amaxk[(size_t)dst * H_HEADS + h], f2key(a));
    }
}

// ex = exp(alpha - amax[dst]); denom[dst] += ex
__global__ void edge_exp_kernel(const int* __restrict__ ei, const unsigned* __restrict__ amaxk,
                                float* __restrict__ alpha, float* __restrict__ denom)
{
    const int i = blockIdx.x * blockDim.x + threadIdx.x;
    if (i >= N_EDGES * H_HEADS) return;
    const int eid = i >> 2;
    const int h   = i & 3;
    const int dst = ei[N_EDGES + eid];
    const float ex = expf(alpha[i] - key2f(amaxk[(size_t)dst * H_HEADS + h]));
    alpha[i] = ex;
    atomicAdd(&denom[(size_t)dst * H_HEADS + h], ex);
}

// out[dst] += (v[src]+e) * ex/denom[dst]
__global__ void edge_msg_kernel(const float* __restrict__ v, const float* __restrict__ edge_attr,
                                const float* __restrict__ We, const float* __restrict__ be,
                                const int* __restrict__ ei, const float* __restrict__ alpha,
                                const float* __restrict__ denom, float* __restrict__ outAgg)
{
    const int eid  = blockIdx.x;
    const int h    = threadIdx.x >> 5;
    const int lane = threadIdx.x & 31;
    const int src = ei[eid];
    const int dst = ei[N_EDGES + eid];
    const float att = alpha[(size_t)eid * H_HEADS + h] /
                      (denom[(size_t)dst * H_HEADS + h] + 1e-16f);
    const int col = h * C_DIM + lane * 4;
    const float4 vv = *(const float4*)(v + (size_t)src * HC + col);
    const float4 ev = edge_feat4(edge_attr + (size_t)eid * EDGE_DIM, We, be, col);
    float* o = outAgg + (size_t)dst * HC + col;
    atomicAdd(o + 0, (vv.x + ev.x) * att);
    atomicAdd(o + 1, (vv.y + ev.y) * att);
    atomicAdd(o + 2, (vv.z + ev.z) * att);
    atomicAdd(o + 3, (vv.w + ev.w) * att);
}

// beta-gated skip: g = sigmoid([out, r, out-r] @ Wbeta + b); out <- g*r + (1-g)*out
__global__ void gate_kernel(float* __restrict__ outAgg, const float* __restrict__ r,
                            const float* __restrict__ Wb, const float* __restrict__ bb)
{
    const int wave = threadIdx.x >> 5;
    const int lane = threadIdx.x & 31;
    const int node = blockIdx.x * 8 + wave;
    if (node >= N_NODES) return;
    const float4* o4 = (const float4*)(outAgg + (size_t)node * HC);
    const float4* r4 = (const float4*)(r + (size_t)node * HC);
    const float4* w0 = (const float4*)(Wb);
    const float4* w1 = (const float4*)(Wb + HC);
    const float4* w2 = (const float4*)(Wb + 2 * HC);
    float d = 0.f;
#pragma unroll
    for (int j = lane; j < HC / 4; j += 32) {
        const float4 o = o4[j], rr = r4[j], a = w0[j], b = w1[j], c = w2[j];
        d += o.x*a.x + rr.x*b.x + (o.x-rr.x)*c.x;
        d += o.y*a.y + rr.y*b.y + (o.y-rr.y)*c.y;
        d += o.z*a.z + rr.z*b.z + (o.z-rr.z)*c.z;
        d += o.w*a.w + rr.w*b.w + (o.w-rr.w)*c.w;
    }
#pragma unroll
    for (int m = 16; m > 0; m >>= 1) d += __shfl_xor(d, m);
    const float g = 1.f / (1.f + expf(-(d + bb[0])));
    float4* ow = (float4*)(outAgg + (size_t)node * HC);
#pragma unroll
    for (int j = lane; j < HC / 4; j += 32) {
        const float4 o = o4[j], rr = r4[j];
        float4 hh;
        hh.x = g * rr.x + (1.f - g) * o.x;
        hh.y = g * rr.y + (1.f - g) * o.y;
        hh.z = g * rr.z + (1.f - g) * o.z;
        hh.w = g * rr.w + (1.f - g) * o.w;
        ow[j] = hh;
    }
}

// per-channel sum / sumsq over N rows
__global__ void bn_stats_kernel(const float* __restrict__ h2, float* __restrict__ sums)
{
    const int c   = threadIdx.x & (OUT_DIM - 1);
    const int sub = threadIdx.x >> 7;
    float s = 0.f, s2 = 0.f;
    for (int row = blockIdx.x * 2 + sub; row < N_NODES; row += gridDim.x * 2) {
        const float v = h2[(size_t)row * OUT_DIM + c];
        s += v; s2 += v * v;
    }
    atomicAdd(&sums[c], s);
    atomicAdd(&sums[OUT_DIM + c], s2);
}

__global__ void bn_apply_kernel(const float* __restrict__ h2, const float* __restrict__ sums,
                                const float* __restrict__ gamma, const float* __restrict__ beta,
                                float* __restrict__ out)
{
    const int i = blockIdx.x * blockDim.x + threadIdx.x;
    if (i >= N_NODES * OUT_DIM) return;
    const int c = i & (OUT_DIM - 1);
    const float invN = 1.f / (float)N_NODES;
    const float mu  = sums[c] * invN;
    const float var = sums[OUT_DIM + c] * invN - mu * mu;
    out[i] = gamma[c] * (h2[i] - mu) * rsqrtf(var + BN_EPS) + beta[c];
}

extern "C" void kernel_launch(void* const* d_in, const int* in_sizes, int n_in,
                              void* d_out, int out_size, void* d_ws, size_t ws_size,
                              hipStream_t stream)
{
    (void)in_sizes; (void)n_in; (void)out_size; (void)ws_size;
    const float* x         = (const float*)d_in[0];
    const float* edge_attr = (const float*)d_in[1];
    const int*   ei        = (const int*)d_in[2];
    // d_in[3] = batch_index (unused by the reference output)
    const float* Wq = (const float*)d_in[4];   const float* bq = (const float*)d_in[5];
    const float* Wk = (const float*)d_in[6];   const float* bk = (const float*)d_in[7];
    const float* Wv = (const float*)d_in[8];   const float* bv = (const float*)d_in[9];
    const float* We = (const float*)d_in[10];  const float* be = (const float*)d_in[11];
    const float* Ws = (const float*)d_in[12];  const float* bs = (const float*)d_in[13];
    const float* Wbeta = (const float*)d_in[14]; const float* bbeta = (const float*)d_in[15];
    const float* Wt = (const float*)d_in[16];  const float* bt = (const float*)d_in[17];
    const float* gamma = (const float*)d_in[18]; const float* beta_bn = (const float*)d_in[19];

    char* ws = (char*)d_ws;
    size_t off = 0;
    auto carve = [&](size_t bytes) -> void* {
        void* p = ws + off;
        off += (bytes + 255) & ~(size_t)255;
        return p;
    };
    float*    q      = (float*)carve((size_t)N_NODES * HC * 4);
    float*    kbuf   = (float*)carve((size_t)N_NODES * HC * 4);
    float*    v      = (float*)carve((size_t)N_NODES * HC * 4);
    float*    r      = (float*)carve((size_t)N_NODES * HC * 4);
    float*    outAgg = (float*)carve((size_t)N_NODES * HC * 4);
    float*    alpha  = (float*)carve((size_t)N_EDGES * H_HEADS * 4);
    unsigned* amaxk  = (unsigned*)carve((size_t)N_NODES * H_HEADS * 4);
    float*    denom  = (float*)carve((size_t)N_NODES * H_HEADS * 4);
    float*    h2     = (float*)carve((size_t)N_NODES * OUT_DIM * 4);
    float*    sums   = (float*)carve((size_t)2 * OUT_DIM * 4);

    hipMemsetAsync(amaxk, 0, (size_t)N_NODES * H_HEADS * 4, stream); // key 0 == -inf floor
    hipMemsetAsync(denom, 0, (size_t)N_NODES * H_HEADS * 4, stream);
    hipMemsetAsync(outAgg, 0, (size_t)N_NODES * HC * 4, stream);
    hipMemsetAsync(sums, 0, (size_t)2 * OUT_DIM * 4, stream);

    const int mblocks = (N_NODES / 16 + 3) / 4;   // 782 blocks of 4 m-tiles
    dim3 gB(HC / 64, mblocks);                    // (8, 782)
    wmma_gemm_bias<F_INDIM, HC, false><<<gB, 128, 0, stream>>>(x, Wq, bq, q,    N_NODES);
    wmma_gemm_bias<F_INDIM, HC, false><<<gB, 128, 0, stream>>>(x, Wk, bk, kbuf, N_NODES);
    wmma_gemm_bias<F_INDIM, HC, false><<<gB, 128, 0, stream>>>(x, Wv, bv, v,    N_NODES);
    wmma_gemm_bias<F_INDIM, HC, false><<<gB, 128, 0, stream>>>(x, Ws, bs, r,    N_NODES);

    edge_alpha_kernel<<<N_EDGES, 128, 0, stream>>>(q, kbuf, edge_attr, We, be, ei, alpha, amaxk);
    edge_exp_kernel<<<(N_EDGES * H_HEADS + 255) / 256, 256, 0, stream>>>(ei, amaxk, alpha, denom);
    edge_msg_kernel<<<N_EDGES, 128, 0, stream>>>(v, edge_attr, We, be, ei, alpha, denom, outAgg);

    gate_kernel<<<(N_NODES + 7) / 8, 256, 0, stream>>>(outAgg, r, Wbeta, bbeta);

    dim3 gT(OUT_DIM / 64, mblocks);               // (2, 782)
    wmma_gemm_bias<HC, OUT_DIM, true><<<gT, 128, 0, stream>>>(outAgg, Wt, bt, h2, N_NODES);

    bn_stats_kernel<<<240, 256, 0, stream>>>(h2, sums);
    bn_apply_kernel<<<(N_NODES * OUT_DIM + 255) / 256, 256, 0, stream>>>(
        h2, sums, gamma, beta_bn, (float*)d_out);
}